// CoordAttention_41721312313859
// MI455X (gfx1250) — compile-verified
//
#include <hip/hip_runtime.h>
#include <math.h>

// ---------------------------------------------------------------------------
// Sizes (from reference)
// ---------------------------------------------------------------------------
#define D_MODEL   64
#define D_INNER   512
#define NHEADS    8
#define HEADDIM   64
#define DSTATE    128
#define DCONV     4
#define CONV_DIM  768          // D_INNER + 2*DSTATE
#define D_IN_PROJ 1288         // 2*D_INNER + 2*DSTATE + NHEADS
#define LSEQ      192          // d1+d2+d3
#define BSZ       2
#define ROWS      (BSZ*LSEQ)   // 384
#define NVOX      4096         // 64*64 tile
#define EPS       1e-5f

typedef float v2f __attribute__((ext_vector_type(2)));
typedef float v8f __attribute__((ext_vector_type(8)));

__device__ __forceinline__ float sigmoidf_(float x){ return 1.0f/(1.0f + __expf(-x)); }
__device__ __forceinline__ float siluf_(float x){ return x * sigmoidf_(x); }

// ---------------------------------------------------------------------------
// Stats + axis-max pass over data.  One block per (b,c,d1) 64x64 tile.
// Produces: block sum/sumsq, a1raw[blk], prow[blk*64+d2], pcol[blk*64+d3].
// ---------------------------------------------------------------------------
__global__ void stats_kernel(const float* __restrict__ data,
                             float* __restrict__ bsum, float* __restrict__ bsq,
                             float* __restrict__ a1raw,
                             float* __restrict__ prow, float* __restrict__ pcol)
{
    const int blk = blockIdx.x;                 // ((b*64+c)*64 + d1)
    const float* tile = data + (size_t)blk * NVOX;
    const int tid = threadIdx.x;                // 256 threads
    __shared__ float lds[256];
    __shared__ float colmax[64];

    // pass 1: thread owns fixed d3 = tid&63, strides over d2
    float sum = 0.f, sq = 0.f, cmax = -__builtin_inff();
#pragma unroll
    for (int i = 0; i < 16; ++i) {
        float v = tile[tid + 256*i];
        sum += v; sq += v*v; cmax = fmaxf(cmax, v);
    }
    lds[tid] = cmax; __syncthreads();
    if (tid < 64) {
        float m = fmaxf(fmaxf(lds[tid], lds[tid+64]), fmaxf(lds[tid+128], lds[tid+192]));
        colmax[tid] = m;
        pcol[blk*64 + tid] = m;                 // partial for a3 (per d3)
    }
    __syncthreads();

    lds[tid] = sum; __syncthreads();
    for (int s = 128; s > 0; s >>= 1) { if (tid < s) lds[tid] += lds[tid+s]; __syncthreads(); }
    if (tid == 0) bsum[blk] = lds[0];
    __syncthreads();
    lds[tid] = sq; __syncthreads();
    for (int s = 128; s > 0; s >>= 1) { if (tid < s) lds[tid] += lds[tid+s]; __syncthreads(); }
    if (tid == 0) bsq[blk] = lds[0];
    __syncthreads();

    // pass 2: row maxes (fixed d2 = tid&63, 16-elem d3 chunk)
    const int d2 = tid & 63, c0 = (tid >> 6) * 16;
    float rmax = -__builtin_inff();
#pragma unroll
    for (int i = 0; i < 16; ++i) rmax = fmaxf(rmax, tile[d2*64 + c0 + i]);
    lds[tid] = rmax; __syncthreads();
    if (tid < 64) {
        float m = fmaxf(fmaxf(lds[tid], lds[tid+64]), fmaxf(lds[tid+128], lds[tid+192]));
        prow[blk*64 + tid] = m;                 // partial for a2 (per d2)
    }
    __syncthreads();

    // a1raw[blk] = full tile max (reduce colmax)
    if (tid < 32) colmax[tid] = fmaxf(colmax[tid], colmax[tid+32]);
    __syncthreads();
    if (tid < 16) colmax[tid] = fmaxf(colmax[tid], colmax[tid+16]);
    __syncthreads();
    if (tid < 8)  colmax[tid] = fmaxf(colmax[tid], colmax[tid+8]);
    __syncthreads();
    if (tid == 0) {
        float m = colmax[0];
        for (int i = 1; i < 8; ++i) m = fmaxf(m, colmax[i]);
        a1raw[blk] = m;
    }
}

// ---------------------------------------------------------------------------
// Per-(b,c) mean / rstd from block partials
// ---------------------------------------------------------------------------
__global__ void finalize_stats(const float* __restrict__ bsum, const float* __restrict__ bsq,
                               float* __restrict__ mu, float* __restrict__ rstd)
{
    const int bc = blockIdx.x; const int tid = threadIdx.x;   // 64 threads
    __shared__ float s1[64], s2[64];
    s1[tid] = bsum[bc*64 + tid];
    s2[tid] = bsq [bc*64 + tid];
    __syncthreads();
    for (int s = 32; s > 0; s >>= 1) {
        if (tid < s) { s1[tid] += s1[tid+s]; s2[tid] += s2[tid+s]; }
        __syncthreads();
    }
    if (tid == 0) {
        const float inv = 1.0f / 262144.0f;      // 64^3
        float m = s1[0] * inv;
        float v = s2[0] * inv - m*m;
        mu[bc] = m;
        rstd[bc] = rsqrtf(v + EPS);
    }
}

// ---------------------------------------------------------------------------
// Reduce a2/a3 partials over d1, normalize all three axis-max arrays, and
// store TRANSPOSED as (b, d, c) so the c1/c2/c3 GEMMs read row-major A.
// ---------------------------------------------------------------------------
__global__ void reduce_axes(const float* __restrict__ a1raw,
                            const float* __restrict__ prow, const float* __restrict__ pcol,
                            const float* __restrict__ mu, const float* __restrict__ rstd,
                            float* __restrict__ a1n, float* __restrict__ a2n, float* __restrict__ a3n)
{
    const int bc = blockIdx.x;                 // b*64+c
    const int b = bc >> 6, c = bc & 63;
    const int d = threadIdx.x;                 // 64 threads
    float m2 = -__builtin_inff(), m3 = -__builtin_inff();
    for (int d1 = 0; d1 < 64; ++d1) {
        m2 = fmaxf(m2, prow[(bc*64 + d1)*64 + d]);
        m3 = fmaxf(m3, pcol[(bc*64 + d1)*64 + d]);
    }
    const float muv = mu[bc], rs = rstd[bc];
    const int o = b*4096 + d*64 + c;           // (b, d, c)
    a1n[o] = (a1raw[bc*64 + d] - muv) * rs;
    a2n[o] = (m2 - muv) * rs;
    a3n[o] = (m3 - muv) * rs;
}

// ---------------------------------------------------------------------------
// Generic fp32 WMMA GEMM:  out[m,n] = sum_k A[m,k] * W[n,k] (+ bias[n])
// One 16x16 output tile per wave, V_WMMA_F32_16X16X4_F32, K in steps of 8
// with two independent accumulator tiles (breaks the WMMA D->C RAW chain).
// OOB lanes use CLAMPED addresses (loads always legal & unconditional; a
// clamped lane only corrupts its own D row/col, which is masked at store).
// Layouts per CDNA5 ISA §7.12.2:
//   A/B: K = reg + 2*(lane/16);  C/D: M = reg + 8*(lane/16), N = lane%16.
// Requires K % 8 == 0 (true for all call sites: 64, 512, 64).
// ---------------------------------------------------------------------------
__global__ void gemm_wmma_f32(const float* __restrict__ A, int lda,
                              const float* __restrict__ W, int ldw,
                              const float* __restrict__ bias,
                              float* __restrict__ C, int ldc,
                              int M, int N, int K)
{
    const int wavesPerBlock = blockDim.x >> 5;
    const int wave = threadIdx.x >> 5;
    const int lane = threadIdx.x & 31;
    const int ntn = (N + 15) >> 4;
    const int ntm = (M + 15) >> 4;
    const int tileId = blockIdx.x * wavesPerBlock + wave;
    if (tileId >= ntm * ntn) return;
    const int tm = tileId / ntn, tn = tileId % ntn;

    const int row = tm*16 + (lane & 15);
    const int col = tn*16 + (lane & 15);
    const int rowC = row < M ? row : (M - 1);   // clamp: load-safe, store-masked
    const int colC = col < N ? col : (N - 1);
    const int khalf = (lane >> 4) * 2;
    const float* Arow = A + (size_t)rowC * lda + khalf;
    const float* Wrow = W + (size_t)colC * ldw + khalf;

    v8f acc0 = {}, acc1 = {};
    for (int kb = 0; kb < K; kb += 8) {
        v2f a0, b0, a1, b1;
        a0.x = Arow[kb];     a0.y = Arow[kb + 1];
        b0.x = Wrow[kb];     b0.y = Wrow[kb + 1];
        a1.x = Arow[kb + 4]; a1.y = Arow[kb + 5];
        b1.x = Wrow[kb + 4]; b1.y = Wrow[kb + 5];
        acc0 = __builtin_amdgcn_wmma_f32_16x16x4_f32(false, a0, false, b0,
                                                     (short)0, acc0, false, false);
        acc1 = __builtin_amdgcn_wmma_f32_16x16x4_f32(false, a1, false, b1,
                                                     (short)0, acc1, false, false);
    }
    const bool colValid = col < N;
    const float bv = (bias && colValid) ? bias[col] : 0.0f;
    const int mbase = tm*16 + 8*(lane >> 4);
#pragma unroll
    for (int v = 0; v < 8; ++v) {
        const int m = mbase + v;
        if (m < M && colValid) C[(size_t)m * ldc + col] = acc0[v] + acc1[v] + bv;
    }
}

// ---------------------------------------------------------------------------
// LayerNorm over 64-wide rows; optional fused residual add (in2).
// ---------------------------------------------------------------------------
__global__ void ln_kernel(const float* __restrict__ in, const float* __restrict__ in2,
                          const float* __restrict__ w, const float* __restrict__ b,
                          float* __restrict__ out)
{
    const int row = blockIdx.x; const int tid = threadIdx.x;   // 64 threads
    __shared__ float red[64];
    float v = in[row*64 + tid];
    if (in2) v += in2[row*64 + tid];
    red[tid] = v; __syncthreads();
    for (int s = 32; s > 0; s >>= 1) { if (tid < s) red[tid] += red[tid+s]; __syncthreads(); }
    const float mu = red[0] * (1.0f/64.0f);
    __syncthreads();
    const float d = v - mu;
    red[tid] = d*d; __syncthreads();
    for (int s = 32; s > 0; s >>= 1) { if (tid < s) red[tid] += red[tid+s]; __syncthreads(); }
    const float var = red[0] * (1.0f/64.0f);
    out[row*64 + tid] = d * rsqrtf(var + EPS) * w[tid] + b[tid];
}

// ---------------------------------------------------------------------------
// Causal depthwise conv (k=4) + SiLU on xBC slice of zxbcdt.
// ---------------------------------------------------------------------------
__global__ void conv_silu_kernel(const float* __restrict__ zx,
                                 const float* __restrict__ conv_w,   // (768,4)
                                 const float* __restrict__ conv_b,
                                 float* __restrict__ xbc)            // (384,768)
{
    const int row = blockIdx.x;               // 0..383
    const int b = row / LSEQ, l = row % LSEQ;
    for (int ch = threadIdx.x; ch < CONV_DIM; ch += blockDim.x) {
        float acc = conv_b[ch];
#pragma unroll
        for (int k = 0; k < DCONV; ++k) {
            const int lt = l + k - (DCONV - 1);
            if (lt >= 0)
                acc += zx[(size_t)(b*LSEQ + lt)*D_IN_PROJ + D_INNER + ch] * conv_w[ch*DCONV + k];
        }
        xbc[(size_t)row*CONV_DIM + ch] = siluf_(acc);
    }
}

// ---------------------------------------------------------------------------
// SSD sequential scan.  One block per (b,h); 256 threads.
// Thread owns p = tid/4, n-chunk of 32 (q = tid&3) -> 32 fp32 state regs.
// ---------------------------------------------------------------------------
__global__ void scan_kernel(const float* __restrict__ zx,      // (384,1288) for dt
                            const float* __restrict__ xbc,     // (384,768)
                            const float* __restrict__ dt_bias, // (8)
                            const float* __restrict__ A_log,   // (8)
                            const float* __restrict__ Dp,      // (8)
                            float* __restrict__ y)             // (384,512)
{
    const int bh = blockIdx.x;               // 0..15
    const int b = bh >> 3, h = bh & 7;
    const int tid = threadIdx.x;             // 256
    const int p = tid >> 2, q = tid & 3, n0 = q*32;
    const float Ah = -__expf(A_log[h]);
    const float Dh = Dp[h];
    const float dtb = dt_bias[h];

    float s[32];
#pragma unroll
    for (int i = 0; i < 32; ++i) s[i] = 0.f;

    __shared__ float Bs[DSTATE], Cs[DSTATE];
    for (int t = 0; t < LSEQ; ++t) {
        const int row = b*LSEQ + t;
        const float* xr = xbc + (size_t)row*CONV_DIM;
        if (tid < 128)      Bs[tid]      = xr[D_INNER + tid];
        else                Cs[tid-128]  = xr[D_INNER + DSTATE + (tid-128)];
        __syncthreads();

        const float dtraw = zx[(size_t)row*D_IN_PROJ + (D_IN_PROJ - NHEADS) + h] + dtb;
        const float dtv = (dtraw > 20.f) ? dtraw : log1pf(__expf(dtraw));  // softplus
        const float da  = __expf(dtv * Ah);
        const float xv  = xr[h*HEADDIM + p];
        const float dtx = dtv * xv;

        float yp = 0.f;
#pragma unroll
        for (int i = 0; i < 32; ++i) {
            s[i] = s[i]*da + dtx * Bs[n0+i];
            yp  += s[i] * Cs[n0+i];
        }
        yp += __shfl_xor(yp, 1);
        yp += __shfl_xor(yp, 2);
        if (q == 0) y[(size_t)row*D_INNER + h*HEADDIM + p] = yp + Dh * xv;
        __syncthreads();
    }
}

// ---------------------------------------------------------------------------
// g = y * silu(z); RMS-norm over 512; * nrm_w
// ---------------------------------------------------------------------------
__global__ void gate_rms_kernel(const float* __restrict__ y, const float* __restrict__ zx,
                                const float* __restrict__ nrm_w, float* __restrict__ g)
{
    const int row = blockIdx.x; const int tid = threadIdx.x;  // 256
    __shared__ float red[256];
    float vals[2]; float ss = 0.f;
#pragma unroll
    for (int i = 0; i < 2; ++i) {
        const int c = tid + i*256;
        const float z = zx[(size_t)row*D_IN_PROJ + c];
        const float gv = y[(size_t)row*D_INNER + c] * siluf_(z);
        vals[i] = gv; ss += gv*gv;
    }
    red[tid] = ss; __syncthreads();
    for (int s = 128; s > 0; s >>= 1) { if (tid < s) red[tid] += red[tid+s]; __syncthreads(); }
    const float r = rsqrtf(red[0] * (1.0f/512.0f) + EPS);
#pragma unroll
    for (int i = 0; i < 2; ++i) {
        const int c = tid + i*256;
        g[(size_t)row*D_INNER + c] = vals[i] * r * nrm_w[c];
    }
}

// ---------------------------------------------------------------------------
// Reverse sequence along L (for the reverse-direction block)
// ---------------------------------------------------------------------------
__global__ void reverse_seq(const float* __restrict__ in, float* __restrict__ out)
{
    const int idx = blockIdx.x * blockDim.x + threadIdx.x;    // 24576
    if (idx >= BSZ*LSEQ*D_MODEL) return;
    const int c = idx & 63, l = (idx >> 6) % LSEQ, b = idx / (LSEQ*D_MODEL);
    out[(size_t)(b*LSEQ + (LSEQ-1-l))*D_MODEL + c] = in[idx];
}

// ca = fwd + reverse(rev); gate = sigmoid(sigmoid(ca))   (as in reference)
__global__ void combine_ca(const float* __restrict__ of, const float* __restrict__ orr,
                           float* __restrict__ gate)
{
    const int idx = blockIdx.x * blockDim.x + threadIdx.x;    // 24576
    if (idx >= BSZ*LSEQ*D_MODEL) return;
    const int c = idx & 63, l = (idx >> 6) % LSEQ, b = idx / (LSEQ*D_MODEL);
    const float v = of[idx] + orr[(size_t)(b*LSEQ + (LSEQ-1-l))*D_MODEL + c];
    gate[idx] = sigmoidf_(sigmoidf_(v));
}

// ---------------------------------------------------------------------------
// Final: out = (data-mu)*rstd * g1[i] * g2[j] * g3[k].  One block per (b,c,i).
// ---------------------------------------------------------------------------
__global__ void final_kernel(const float* __restrict__ data, const float* __restrict__ gate,
                             const float* __restrict__ mu, const float* __restrict__ rstd,
                             float* __restrict__ out)
{
    const int blk = blockIdx.x;               // ((b*64+c)*64 + i)
    const int i = blk & 63, bc = blk >> 6;
    const int b = bc >> 6, c = bc & 63;
    __shared__ float G2[64], G3[64];
    const int tid = threadIdx.x;              // 256
    if (tid < 64)       G2[tid]    = gate[(size_t)(b*LSEQ +  64 + tid)*D_MODEL + c];
    else if (tid < 128) G3[tid-64] = gate[(size_t)(b*LSEQ + 128 + (tid-64))*D_MODEL + c];
    __syncthreads();
    const float g1 = gate[(size_t)(b*LSEQ + i)*D_MODEL + c];
    const float muv = mu[bc], rs = rstd[bc];
    const float* tile = data + (size_t)blk * NVOX;
    float* otile = out + (size_t)blk * NVOX;
#pragma unroll
    for (int it = 0; it < 16; ++it) {
        const int idx = tid + 256*it;
        const int j = idx >> 6, k = idx & 63;
        const float xv = (tile[idx] - muv) * rs;
        otile[idx] = xv * g1 * G2[j] * G3[k];
    }
}

// ---------------------------------------------------------------------------
// Host side
// ---------------------------------------------------------------------------
struct MambaParams {
    const float *ln1_w, *ln1_b, *in_w, *conv_w, *conv_b, *dt_bias, *A_log,
                *D, *nrm_w, *out_w, *ln2_w, *ln2_b, *mlp_w, *mlp_b;
};

struct DirBufs {
    float *u_ln, *zx, *xbc, *y, *g, *mix, *h2, *outdir;
};

static void launch_gemm(const float* A, int lda, const float* W, int ldw,
                        const float* bias, float* C, int ldc,
                        int M, int N, int K, hipStream_t stream)
{
    const int ntiles = ((M + 15)/16) * ((N + 15)/16);
    const int wavesPerBlock = 4;                 // 128-thread blocks
    const int grid = (ntiles + wavesPerBlock - 1) / wavesPerBlock;
    gemm_wmma_f32<<<grid, wavesPerBlock*32, 0, stream>>>(A, lda, W, ldw, bias, C, ldc, M, N, K);
}

static void run_mamba_block(const float* seq, const DirBufs& bf, const MambaParams& p,
                            hipStream_t stream)
{
    // ln1
    ln_kernel<<<ROWS, 64, 0, stream>>>(seq, nullptr, p.ln1_w, p.ln1_b, bf.u_ln);
    // in-proj (384x64)x(1288x64)^T -> (384x1288)
    launch_gemm(bf.u_ln, D_MODEL, p.in_w, D_MODEL, nullptr, bf.zx, D_IN_PROJ,
                ROWS, D_IN_PROJ, D_MODEL, stream);
    // conv + silu
    conv_silu_kernel<<<ROWS, 256, 0, stream>>>(bf.zx, p.conv_w, p.conv_b, bf.xbc);
    // SSD scan
    scan_kernel<<<BSZ*NHEADS, 256, 0, stream>>>(bf.zx, bf.xbc, p.dt_bias, p.A_log, p.D, bf.y);
    // gating + RMS norm
    gate_rms_kernel<<<ROWS, 256, 0, stream>>>(bf.y, bf.zx, p.nrm_w, bf.g);
    // out-proj (384x512)x(64x512)^T -> (384x64)
    launch_gemm(bf.g, D_INNER, p.out_w, D_INNER, nullptr, bf.mix, D_MODEL,
                ROWS, D_MODEL, D_INNER, stream);
    // ln2 over (mix + seq)
    ln_kernel<<<ROWS, 64, 0, stream>>>(bf.mix, seq, p.ln2_w, p.ln2_b, bf.h2);
    // mlp (384x64)x(64x64)^T + b -> (384x64)
    launch_gemm(bf.h2, D_MODEL, p.mlp_w, D_MODEL, p.mlp_b, bf.outdir, D_MODEL,
                ROWS, D_MODEL, D_MODEL, stream);
}

extern "C" void kernel_launch(void* const* d_in, const int* in_sizes, int n_in,
                              void* d_out, int out_size, void* d_ws, size_t ws_size,
                              hipStream_t stream)
{
    (void)in_sizes; (void)n_in; (void)out_size; (void)ws_size;
    const float* data = (const float*)d_in[0];
    const float* w1 = (const float*)d_in[1]; const float* b1 = (const float*)d_in[2];
    const float* w2 = (const float*)d_in[3]; const float* b2 = (const float*)d_in[4];
    const float* w3 = (const float*)d_in[5]; const float* b3 = (const float*)d_in[6];

    auto load_params = [&](int base) {
        MambaParams p;
        p.ln1_w  = (const float*)d_in[base+0];  p.ln1_b  = (const float*)d_in[base+1];
        p.in_w   = (const float*)d_in[base+2];  p.conv_w = (const float*)d_in[base+3];
        p.conv_b = (const float*)d_in[base+4];  p.dt_bias= (const float*)d_in[base+5];
        p.A_log  = (const float*)d_in[base+6];  p.D      = (const float*)d_in[base+7];
        p.nrm_w  = (const float*)d_in[base+8];  p.out_w  = (const float*)d_in[base+9];
        p.ln2_w  = (const float*)d_in[base+10]; p.ln2_b  = (const float*)d_in[base+11];
        p.mlp_w  = (const float*)d_in[base+12]; p.mlp_b  = (const float*)d_in[base+13];
        return p;
    };
    const MambaParams pf = load_params(7);
    const MambaParams pr = load_params(21);

    // ---- workspace layout (floats) ----
    float* ws = (float*)d_ws;
    size_t o = 0;
    auto alloc = [&](size_t n) { float* p = ws + o; o += n; return p; };
    float* bsum  = alloc(8192);
    float* bsq   = alloc(8192);
    float* a1raw = alloc(8192);
    float* prow  = alloc(524288);
    float* pcol  = alloc(524288);
    float* mu    = alloc(128);
    float* rstd  = alloc(128);
    float* a1n   = alloc(8192);
    float* a2n   = alloc(8192);
    float* a3n   = alloc(8192);
    float* ca_in = alloc((size_t)BSZ*LSEQ*D_MODEL);
    float* seq_r = alloc((size_t)BSZ*LSEQ*D_MODEL);
    float* gate  = alloc((size_t)BSZ*LSEQ*D_MODEL);
    DirBufs bufs[2];
    for (int d = 0; d < 2; ++d) {
        bufs[d].u_ln   = alloc((size_t)ROWS*D_MODEL);
        bufs[d].zx     = alloc((size_t)ROWS*D_IN_PROJ);
        bufs[d].xbc    = alloc((size_t)ROWS*CONV_DIM);
        bufs[d].y      = alloc((size_t)ROWS*D_INNER);
        bufs[d].g      = alloc((size_t)ROWS*D_INNER);
        bufs[d].mix    = alloc((size_t)ROWS*D_MODEL);
        bufs[d].h2     = alloc((size_t)ROWS*D_MODEL);
        bufs[d].outdir = alloc((size_t)ROWS*D_MODEL);
    }

    // ---- phase 1: big-tensor stats + axis maxes ----
    stats_kernel<<<8192, 256, 0, stream>>>(data, bsum, bsq, a1raw, prow, pcol);
    finalize_stats<<<128, 64, 0, stream>>>(bsum, bsq, mu, rstd);
    reduce_axes<<<128, 64, 0, stream>>>(a1raw, prow, pcol, mu, rstd, a1n, a2n, a3n);

    // ---- phase 2: c1/c2/c3 projections -> ca_in (B,192,64) ----
    for (int b = 0; b < BSZ; ++b) {
        launch_gemm(a1n + (size_t)b*4096, 64, w1, 64, b1,
                    ca_in + (size_t)(b*LSEQ +   0)*64, 64, 64, 64, 64, stream);
        launch_gemm(a2n + (size_t)b*4096, 64, w2, 64, b2,
                    ca_in + (size_t)(b*LSEQ +  64)*64, 64, 64, 64, 64, stream);
        launch_gemm(a3n + (size_t)b*4096, 64, w3, 64, b3,
                    ca_in + (size_t)(b*LSEQ + 128)*64, 64, 64, 64, 64, stream);
    }

    // ---- phase 3: forward + reverse Mamba2 blocks ----
    const int nelem = BSZ*LSEQ*D_MODEL;
    reverse_seq<<<(nelem + 255)/256, 256, 0, stream>>>(ca_in, seq_r);
    run_mamba_block(ca_in, bufs[0], pf, stream);
    run_mamba_block(seq_r, bufs[1], pr, stream);

    // ---- phase 4: combine, gates, final elementwise ----
    combine_ca<<<(nelem + 255)/256, 256, 0, stream>>>(bufs[0].outdir, bufs[1].outdir, gate);
    final_kernel<<<8192, 256, 0, stream>>>(data, gate, mu, rstd, (float*)d_out);
}